// MySoftBCELoss_68547678044206
// MI455X (gfx1250) — compile-verified
//
#include <hip/hip_runtime.h>
#include <hip/hip_bf16.h>

#define EPS 1e-7f
#define NCLS 64

typedef __attribute__((ext_vector_type(2))) float f2;
typedef __attribute__((ext_vector_type(8))) float v8f;

// Kernel 1: one wave32 per row (grid-strided).
//  - lane l loads target[row, 2l..2l+1] (coalesced 256B/row, non-temporal)
//  - butterfly (value,index) argmax with lowest-index tie-break
//  - lane 0 gathers logits[row,0]  -> log(1 - p0) term
//  - lane 16 gathers logits[row,L] -> t_L * log(pL) term   (parallel scattered loads)
//  - per-wave accumulators, fixed-order per-block partial into d_ws
__global__ void __launch_bounds__(256)
soft_bce_main(const float* __restrict__ logits,
              const float* __restrict__ target,
              float* __restrict__ partials, int nrows) {
    const int lane = threadIdx.x & 31;
    const int wib  = threadIdx.x >> 5;
    const int wpb  = blockDim.x >> 5;
    const int gw   = blockIdx.x * wpb + wib;
    const int nw   = gridDim.x * wpb;

    float acc = 0.0f;
    for (int row = gw; row < nrows; row += nw) {
        const f2* trow = (const f2*)(target + (size_t)row * NCLS);
        f2 tv = __builtin_nontemporal_load(trow + lane);

        // local argmax over the two elements this lane owns (tie -> lower index)
        float bv; int bi;
        if (tv.y > tv.x) { bv = tv.y; bi = 2 * lane + 1; }
        else             { bv = tv.x; bi = 2 * lane;     }

        // wave32 butterfly argmax, lowest-index-wins on ties (matches jnp.argmax)
        #pragma unroll
        for (int off = 16; off > 0; off >>= 1) {
            float ov = __shfl_xor(bv, off);
            int   oi = __shfl_xor(bi, off);
            if (ov > bv || (ov == bv && oi < bi)) { bv = ov; bi = oi; }
        }
        // all lanes now agree: bv = target[row, L] (== t_L), bi = L (>= 1)

        const float* lrow = logits + (size_t)row * NCLS;
        if (lane == 0) {
            float x0 = __builtin_nontemporal_load(lrow);
            float p0 = 1.0f / (1.0f + expf(-x0));
            p0 = fminf(fmaxf(p0, EPS), 1.0f - EPS);
            acc += logf(1.0f - p0);
        } else if (lane == 16) {
            float xl = __builtin_nontemporal_load(lrow + bi);
            float pl = 1.0f / (1.0f + expf(-xl));
            pl = fminf(fmaxf(pl, EPS), 1.0f - EPS);
            acc += bv * logf(pl);
        }
    }

    // wave sum (only lanes 0/16 carry nonzero, butterfly is cheap & deterministic)
    #pragma unroll
    for (int off = 16; off > 0; off >>= 1)
        acc += __shfl_xor(acc, off);

    __shared__ float smem[32];
    if (lane == 0) smem[wib] = acc;
    __syncthreads();
    if (threadIdx.x == 0) {
        float s = 0.0f;
        for (int i = 0; i < wpb; ++i) s += smem[i];   // fixed order -> deterministic
        partials[blockIdx.x] = s;
    }
}

// Kernel 2: single wave32. Each lane serially folds a strided slice of the
// block partials (fixed order), then the 32 lane accumulators are summed with
// one V_WMMA_F32_16X16X4_F32 against an all-ones B matrix:
//   A (16x4 layout): VGPR0 lane m   = A[m][0] = acc(lane m)
//                    VGPR0 lane m+16= A[m][2] = acc(lane m+16), A[.][1]=A[.][3]=0
//   D[m][n] = acc(m) + acc(m+16) for every column n.
//   lane0 holds D[0..7][0] in its 8 result VGPRs, lane16 holds D[8..15][0];
//   summing those 8 regs per lane and adding lane0+lane16 gives the 32-lane total.
__global__ void soft_bce_final(const float* __restrict__ partials, int n,
                               float* __restrict__ out, float invB) {
    const int lane = threadIdx.x;   // exactly 32 threads, EXEC all ones
    float a = 0.0f;
    for (int i = lane; i < n; i += 32) a += partials[i];

    f2 A;  A.x = a;    A.y = 0.0f;
    f2 Bm; Bm.x = 1.0f; Bm.y = 1.0f;
    v8f Cv = {};
    v8f D = __builtin_amdgcn_wmma_f32_16x16x4_f32(
        /*neg_a=*/false, A, /*neg_b=*/false, Bm,
        /*c_mod=*/(short)0, Cv, /*reuse_a=*/false, /*reuse_b=*/false);

    float s = D[0] + D[1] + D[2] + D[3] + D[4] + D[5] + D[6] + D[7];
    s += __shfl_xor(s, 16);          // lane0 part + lane16 part
    if (lane == 0) out[0] = -(s * invB);
}

extern "C" void kernel_launch(void* const* d_in, const int* in_sizes, int n_in,
                              void* d_out, int out_size, void* d_ws, size_t ws_size,
                              hipStream_t stream) {
    const float* logits = (const float*)d_in[0];
    const float* target = (const float*)d_in[1];
    float* out      = (float*)d_out;
    float* partials = (float*)d_ws;

    const int nrows = in_sizes[0] / NCLS;   // 524288

    int blocks = 2048;                       // 16384 waves -> 32 rows/wave
    size_t cap = ws_size / sizeof(float);
    if ((size_t)blocks > cap) blocks = (int)(cap ? cap : 1);

    soft_bce_main<<<blocks, 256, 0, stream>>>(logits, target, partials, nrows);
    soft_bce_final<<<1, 32, 0, stream>>>(partials, blocks, out,
                                         1.0f / (float)nrows);
}